// MFFHistoNet_46428596470447
// MI455X (gfx1250) — compile-verified
//
#include <hip/hip_runtime.h>

#define IMG_H 224
#define IMG_W 224
#define NPIX (IMG_H * IMG_W)
#define BATCH 16
#define KBIG 150528      // 3*224*224
#define SPLITK 128
#define CHUNK 1176       // 150528 / 128, multiple of 4

typedef __attribute__((ext_vector_type(2))) float v2f;
typedef __attribute__((ext_vector_type(8))) float v8f;

// ---------------------------------------------------------------- utilities
__device__ inline float block_reduce_256(float v, float* red) {
  int t = threadIdx.x;
  red[t] = v; __syncthreads();
#pragma unroll
  for (int s = 128; s > 0; s >>= 1) {
    if (t < s) red[t] += red[t + s];
    __syncthreads();
  }
  float r = red[0];
  __syncthreads();
  return r;
}

__global__ void k_zero_u32(unsigned* p, int n) {
  int i = blockIdx.x * blockDim.x + threadIdx.x;
  if (i < n) p[i] = 0u;
}

// ------------------------------------------------- gray = rgb2gray(denorm(x))
__global__ void k_gray(const float* __restrict__ x, float* __restrict__ gray) {
  int idx = blockIdx.x * blockDim.x + threadIdx.x;
  if (idx >= BATCH * NPIX) return;
  int b = idx / NPIX, p = idx % NPIX;
  const float mean_[3] = {0.485f, 0.456f, 0.406f};
  const float std_[3]  = {0.229f, 0.224f, 0.225f};
  const float gw_[3]   = {0.2125f, 0.7154f, 0.0721f};
  float s = 0.0f;
#pragma unroll
  for (int c = 0; c < 3; ++c) {
    float v = x[((size_t)(b * 3 + c)) * NPIX + p] * std_[c] + mean_[c];
    v = fminf(fmaxf(v, 0.0f), 1.0f);
    s += gw_[c] * v;
  }
  gray[idx] = s;
}

// ------------------------------------------------- 32x32 avg pool -> [16,147]
__global__ void k_pool(const float* __restrict__ x, float* __restrict__ pooled) {
  __shared__ float red[256];
  int o = blockIdx.x;                 // 16*147 blocks
  int b = o / 147, f = o % 147;
  int c = f / 49, ph = (f / 7) % 7, pw = f % 7;
  const float* base = x + (((size_t)(b * 3 + c)) * IMG_H + ph * 32) * IMG_W + pw * 32;
  float s = 0.0f;
  for (int t = threadIdx.x; t < 1024; t += 256) {
    int i = t >> 5, j = t & 31;
    s += base[i * IMG_W + j];
  }
  float tot = block_reduce_256(s, red);
  if (threadIdx.x == 0) pooled[b * 147 + f] = tot * (1.0f / 1024.0f);
}

// -------------------------------------- generic C[16,N] = act(A[16,K]@W + b)
// one wave per 16-wide N tile, f32 WMMA 16x16x4.
// Branch-free: main loop fully unguarded (K rounded down to x4); over-range N
// columns load a clamped column (their accumulator cols are never stored);
// one wave-uniform tail step handles K%4 with value-selects, not branches.
__global__ void k_gemm16(const float* __restrict__ A, int lda,
                         const float* __restrict__ Wm, int ldw,
                         const float* __restrict__ bias,
                         float* __restrict__ C, int ldc,
                         int K, int N, int relu) {
  const int n0 = blockIdx.x * 16;
  const int l = threadIdx.x;
  const int half = l >> 4;
  const int m = l & 15;
  const int ncol = n0 + m;
  const bool nv = (ncol < N);
  const int ncl = nv ? ncol : (N - 1);           // clamped load column
  const float* Ar = A + (size_t)m * lda + 2 * half;
  const float* Wc = Wm + (size_t)(2 * half) * ldw + ncl;
  const int K4 = K & ~3;
  v8f acc = {};
  for (int k = 0; k < K4; k += 4) {
    v2f a, b;
    a.x = Ar[k];
    a.y = Ar[k + 1];
    b.x = Wc[(size_t)k * ldw];
    b.y = Wc[(size_t)k * ldw + ldw];
    acc = __builtin_amdgcn_wmma_f32_16x16x4_f32(false, a, false, b, (short)0, acc, false, false);
  }
  if (K4 < K) {                                   // wave-uniform tail
    const int ka = K4 + 2 * half;
    const int i0 = (ka < K) ? ka : (K - 1);
    const int i1 = (ka + 1 < K) ? (ka + 1) : (K - 1);
    v2f a, b;
    a.x = A[(size_t)m * lda + i0];
    a.y = A[(size_t)m * lda + i1];
    b.x = Wm[(size_t)i0 * ldw + ncl];
    b.y = Wm[(size_t)i1 * ldw + ncl];
    a.x = (ka < K) ? a.x : 0.0f;
    a.y = (ka + 1 < K) ? a.y : 0.0f;
    b.x = (ka < K) ? b.x : 0.0f;
    b.y = (ka + 1 < K) ? b.y : 0.0f;
    acc = __builtin_amdgcn_wmma_f32_16x16x4_f32(false, a, false, b, (short)0, acc, false, false);
  }
  if (nv) {
    const float bb = bias ? bias[ncol] : 0.0f;
#pragma unroll
    for (int i = 0; i < 8; ++i) {
      const int row = 8 * half + i;
      float v = acc[i] + bb;
      if (relu) v = fmaxf(v, 0.0f);
      C[(size_t)row * ldc + ncol] = v;
    }
  }
}

// -------------------- big GEMM: split-K partials, grid (N/16, SPLITK), 32 thr
// This is the 308 MB W1 stream; vector A loads + unroll-4 + W prefetch to
// keep enough loads in flight per wave to ride the 23.3 TB/s HBM roof.
__global__ void k_gemm16_splitk(const float* __restrict__ A, int lda,
                                const float* __restrict__ Wm, int ldw,
                                float* __restrict__ partial, int N) {
  const int n0 = blockIdx.x * 16;
  const int cId = blockIdx.y;
  const int l = threadIdx.x;
  const int half = l >> 4;
  const int m = l & 15;
  const int ncol = n0 + m;
  const float* Ar = A + (size_t)m * lda + cId * CHUNK + 2 * half;   // 8B aligned
  const float* Wc = Wm + ((size_t)cId * CHUNK + 2 * half) * ldw + ncol;
  v8f acc = {};
#pragma unroll 4
  for (int k = 0; k < CHUNK; k += 4) {
    __builtin_prefetch(Wc + (size_t)(k + 64) * ldw, 0, 1);
    v2f a = *(const v2f*)(Ar + k);
    v2f b;
    b.x = Wc[(size_t)k * ldw];
    b.y = Wc[(size_t)k * ldw + ldw];
    acc = __builtin_amdgcn_wmma_f32_16x16x4_f32(false, a, false, b, (short)0, acc, false, false);
  }
#pragma unroll
  for (int i = 0; i < 8; ++i) {
    const int row = 8 * half + i;
    partial[((size_t)cId * 16 + row) * N + ncol] = acc[i];
  }
}

__global__ void k_reduce_splitk(const float* __restrict__ partial,
                                const float* __restrict__ bias,
                                float* __restrict__ out, int N, int relu) {
  int idx = blockIdx.x * blockDim.x + threadIdx.x;
  if (idx >= 16 * N) return;
  int m = idx / N, n = idx % N;
  float s = bias[n];
  for (int c = 0; c < SPLITK; ++c) s += partial[((size_t)c * 16 + m) * N + n];
  if (relu) s = fmaxf(s, 0.0f);
  out[idx] = s;
}

// ------------------------------------------------------------- GLCM histogram
__global__ void k_glcm_hist(const float* __restrict__ gray, unsigned* __restrict__ counts) {
  int b = blockIdx.y;
  int idx = blockIdx.x * blockDim.x + threadIdx.x;
  if (idx >= IMG_H * (IMG_W - 1)) return;
  int r = idx / (IMG_W - 1), c = idx % (IMG_W - 1);
  const float* g = gray + (size_t)b * NPIX;
  int gi = (int)(g[r * IMG_W + c] * 255.0f);
  int gj = (int)(g[r * IMG_W + c + 1] * 255.0f);
  gi = min(max(gi, 0), 255);
  gj = min(max(gj, 0), 255);
  atomicAdd(&counts[(size_t)b * 65536 + gi * 256 + gj], 1u);
}

// ------------------------------------------------- GLCM -> 6 features (tex38)
__global__ void k_glcm_feats(const unsigned* __restrict__ counts, float* __restrict__ tex38) {
  __shared__ float red[256];
  int b = blockIdx.x;
  const unsigned* C = counts + (size_t)b * 65536;
  float S = 0, Sc = 0, Sd = 0, Sh = 0, Sq2 = 0, Si = 0, Sj = 0, Sii = 0, Sjj = 0, Sij = 0;
  for (int t = threadIdx.x; t < 65536; t += 256) {
    int i = t >> 8, j = t & 255;
    float q = (float)C[t] + (float)C[(j << 8) | i];
    if (q != 0.0f) {
      float d = (float)(i - j);
      float fi = (float)i, fj = (float)j;
      S += q; Sc += q * d * d; Sd += q * fabsf(d); Sh += q / (1.0f + d * d);
      Sq2 += q * q; Si += q * fi; Sj += q * fj;
      Sii += q * fi * fi; Sjj += q * fj * fj; Sij += q * fi * fj;
    }
  }
  S = block_reduce_256(S, red);   Sc = block_reduce_256(Sc, red);
  Sd = block_reduce_256(Sd, red); Sh = block_reduce_256(Sh, red);
  Sq2 = block_reduce_256(Sq2, red); Si = block_reduce_256(Si, red);
  Sj = block_reduce_256(Sj, red); Sii = block_reduce_256(Sii, red);
  Sjj = block_reduce_256(Sjj, red); Sij = block_reduce_256(Sij, red);
  if (threadIdx.x == 0) {
    float inv = 1.0f / S;
    float contrast = Sc * inv, dissim = Sd * inv, homog = Sh * inv;
    float asmv = Sq2 * inv * inv;
    float energy = sqrtf(asmv);
    float mui = Si * inv, muj = Sj * inv;
    float vari = Sii * inv - mui * mui, varj = Sjj * inv - muj * muj;
    float cov = Sij * inv - mui * muj;
    float den = sqrtf(vari * varj);
    float corr = (den < 1e-15f) ? 1.0f : cov / fmaxf(den, 1e-15f);
    float* o = tex38 + b * 38;
    o[0] = contrast; o[1] = dissim; o[2] = homog; o[3] = energy; o[4] = corr; o[5] = asmv;
  }
}

// ------------------------------------------------------------- LBP histogram
__device__ inline float lbp_at(const float* g, float rf, float cf) {
  int ri = (int)rf, ci = (int)cf;           // rf/cf already integral (floor)
  bool ok = (ri >= 0) && (ri < IMG_H) && (ci >= 0) && (ci < IMG_W);
  int rc = min(max(ri, 0), IMG_H - 1), cc = min(max(ci, 0), IMG_W - 1);
  return ok ? g[rc * IMG_W + cc] : 0.0f;
}

__global__ void k_lbp(const float* __restrict__ gray, unsigned* __restrict__ hist) {
  int b = blockIdx.y;
  int pix = blockIdx.x * blockDim.x + threadIdx.x;
  if (pix >= NPIX) return;
  int r = pix / IMG_W, c = pix % IMG_W;
  const float* g = gray + (size_t)b * NPIX;
  float center = g[pix];
  int bits[24];
  int sum = 0;
#pragma unroll 1
  for (int p = 0; p < 24; ++p) {
    float ang = 6.2831853071795864f * (float)p / 24.0f;
    float sr = (float)r - 3.0f * sinf(ang);
    float sc = (float)c + 3.0f * cosf(ang);
    float r0 = floorf(sr), c0 = floorf(sc);
    float dr = sr - r0, dc = sc - c0;
    float v = (1.0f - dr) * (1.0f - dc) * lbp_at(g, r0, c0)
            + (1.0f - dr) * dc          * lbp_at(g, r0, c0 + 1.0f)
            + dr * (1.0f - dc)          * lbp_at(g, r0 + 1.0f, c0)
            + dr * dc                    * lbp_at(g, r0 + 1.0f, c0 + 1.0f);
    bits[p] = (v - center >= 0.0f) ? 1 : 0;
    sum += bits[p];
  }
  int changes = 0;
#pragma unroll
  for (int p = 0; p < 24; ++p) changes += bits[p] ^ bits[(p + 1) % 24];
  int val = (changes <= 2) ? sum : 25;
  atomicAdd(&hist[b * 32 + val], 1u);
}

// --------------------------------------------- Gabor: sum f, sum f^2 per (b,λ)
__global__ void k_gabor(const float* __restrict__ gray, float* __restrict__ sums) {
  __shared__ float kk[441];
  __shared__ float red[256];
  int b = blockIdx.x, li = blockIdx.y;
  float lam = (li == 0) ? 0.1f : ((li == 1) ? 0.5f : 0.9f);
  for (int t = threadIdx.x; t < 441; t += 256) {
    float yy = (float)(t / 21 - 10), xx = (float)(t % 21 - 10);
    kk[t] = expf(-(xx * xx + 0.25f * yy * yy) * (1.0f / 50.0f))
          * cosf(6.2831853071795864f * xx / lam);
  }
  __syncthreads();
  const float* g = gray + (size_t)b * NPIX;
  float sf = 0.0f, sf2 = 0.0f;
  for (int pix = threadIdx.x; pix < NPIX; pix += 256) {
    int r = pix / IMG_W, c = pix % IMG_W;
    float f = 0.0f;
    for (int dy = -10; dy <= 10; ++dy) {
      int rr = r + dy;
      rr = (rr < 0) ? -rr : ((rr > IMG_H - 1) ? 2 * (IMG_H - 1) - rr : rr);
      const float* row = g + rr * IMG_W;
      const float* kr = kk + (dy + 10) * 21;
      for (int dx = -10; dx <= 10; ++dx) {
        int cc = c + dx;
        cc = (cc < 0) ? -cc : ((cc > IMG_W - 1) ? 2 * (IMG_W - 1) - cc : cc);
        f += kr[dx + 10] * row[cc];
      }
    }
    sf += f; sf2 += f * f;
  }
  sf = block_reduce_256(sf, red);
  sf2 = block_reduce_256(sf2, red);
  if (threadIdx.x == 0) {
    sums[(b * 3 + li) * 2 + 0] = sf;
    sums[(b * 3 + li) * 2 + 1] = sf2;
  }
}

// --------------------------------------- assemble tex38 (lbp hist + gabor m/v)
__global__ void k_tex_assemble(const unsigned* __restrict__ lbph,
                               const float* __restrict__ gsums,
                               float* __restrict__ tex38) {
  int b = blockIdx.x, t = threadIdx.x;
  if (t < 26) tex38[b * 38 + 6 + t] = (float)lbph[b * 32 + t] / (50176.0f + 1e-6f);
  if (t < 3) {
    float sf = gsums[(b * 3 + t) * 2], sf2 = gsums[(b * 3 + t) * 2 + 1];
    float m = sf * (1.0f / 50176.0f);
    float var = sf2 * (1.0f / 50176.0f) - m * m;
    tex38[b * 38 + 32 + 2 * t] = m;
    tex38[b * 38 + 33 + 2 * t] = var;
  }
}

// ---------------------------------------------------- concat [cnn|qtn|texf]
__global__ void k_comb(const float* __restrict__ cnn, const float* __restrict__ qtn,
                       const float* __restrict__ texf, float* __restrict__ comb) {
  int b = blockIdx.x, t = threadIdx.x;
  if (t < 5) comb[b * 261 + t] = cnn[b * 5 + t];
  else if (t < 133) comb[b * 261 + t] = qtn[b * 128 + (t - 5)];
  else if (t < 261) comb[b * 261 + t] = texf[b * 128 + (t - 133)];
}

// ============================================================================
extern "C" void kernel_launch(void* const* d_in, const int* in_sizes, int n_in,
                              void* d_out, int out_size, void* d_ws, size_t ws_size,
                              hipStream_t stream) {
  const float* x      = (const float*)d_in[0];
  const float* W_back = (const float*)d_in[1];
  const float* b_back = (const float*)d_in[2];
  const float* W_cls  = (const float*)d_in[3];
  const float* b_cls  = (const float*)d_in[4];
  const float* W1     = (const float*)d_in[5];
  const float* b1     = (const float*)d_in[6];
  const float* Wt1    = (const float*)d_in[7];
  const float* bt1    = (const float*)d_in[8];
  const float* Wt2    = (const float*)d_in[9];
  const float* bt2    = (const float*)d_in[10];
  const float* W2     = (const float*)d_in[11];
  const float* b2     = (const float*)d_in[12];
  const float* W_tex  = (const float*)d_in[13];
  const float* b_tex  = (const float*)d_in[14];
  const float* W_fus  = (const float*)d_in[15];
  const float* b_fus  = (const float*)d_in[16];
  const float* W_fin  = (const float*)d_in[17];
  const float* b_fin  = (const float*)d_in[18];
  float* out = (float*)d_out;

  // ---- bump allocator over workspace
  char* ws = (char*)d_ws;
  size_t off = 0;
  auto alloc = [&](size_t bytes) -> void* {
    void* p = ws + off;
    off = (off + bytes + 255) & ~(size_t)255;
    return p;
  };
  float*    gray    = (float*)alloc((size_t)BATCH * NPIX * 4);          // 3.2 MB
  unsigned* glcm    = (unsigned*)alloc((size_t)BATCH * 65536 * 4);      // 4 MB
  unsigned* lbph    = (unsigned*)alloc((size_t)BATCH * 32 * 4);
  float*    pooled  = (float*)alloc(2352 * 4);
  float*    hb      = (float*)alloc(16 * 1024 * 4);
  float*    cnn     = (float*)alloc(80 * 4);
  float*    partial = (float*)alloc((size_t)SPLITK * 16 * 512 * 4);     // 4 MB
  float*    h1      = (float*)alloc(16 * 512 * 4);
  float*    h2      = (float*)alloc(16 * 512 * 4);
  float*    h3      = (float*)alloc(16 * 512 * 4);
  float*    qtn     = (float*)alloc(16 * 128 * 4);
  float*    gsums   = (float*)alloc(16 * 3 * 2 * 4);
  float*    tex38   = (float*)alloc(16 * 38 * 4);
  float*    texf    = (float*)alloc(16 * 128 * 4);
  float*    comb    = (float*)alloc(16 * 261 * 4);
  float*    fused   = (float*)alloc(16 * 256 * 4);

  // ---- zero histograms (integer atomics -> deterministic)
  k_zero_u32<<<(BATCH * 65536 + 255) / 256, 256, 0, stream>>>(glcm, BATCH * 65536);
  k_zero_u32<<<(BATCH * 32 + 255) / 256, 256, 0, stream>>>(lbph, BATCH * 32);

  // ---- gray image (feeds all texture kernels)
  k_gray<<<(BATCH * NPIX + 255) / 256, 256, 0, stream>>>(x, gray);

  // ---- QTN branch first: the big split-K WMMA GEMM is the HBM-bound tail,
  //      start it as early as possible.
  k_gemm16_splitk<<<dim3(32, SPLITK), 32, 0, stream>>>(x, KBIG, W1, 512, partial, 512);
  k_reduce_splitk<<<(16 * 512 + 255) / 256, 256, 0, stream>>>(partial, b1, h1, 512, 1);
  k_gemm16<<<32, 32, 0, stream>>>(h1, 512, Wt1, 512, bt1, h2, 512, 512, 512, 1);
  k_gemm16<<<32, 32, 0, stream>>>(h2, 512, Wt2, 512, bt2, h3, 512, 512, 512, 0);
  k_gemm16<<<8, 32, 0, stream>>>(h3, 512, W2, 128, b2, qtn, 128, 512, 128, 0);

  // ---- CNN branch
  k_pool<<<BATCH * 147, 256, 0, stream>>>(x, pooled);
  k_gemm16<<<64, 32, 0, stream>>>(pooled, 147, W_back, 1024, b_back, hb, 1024, 147, 1024, 1);
  k_gemm16<<<1, 32, 0, stream>>>(hb, 1024, W_cls, 5, b_cls, cnn, 5, 1024, 5, 0);

  // ---- texture branch
  k_glcm_hist<<<dim3(196, BATCH), 256, 0, stream>>>(gray, glcm);
  k_glcm_feats<<<BATCH, 256, 0, stream>>>(glcm, tex38);
  k_lbp<<<dim3(196, BATCH), 256, 0, stream>>>(gray, lbph);
  k_gabor<<<dim3(BATCH, 3), 256, 0, stream>>>(gray, gsums);
  k_tex_assemble<<<BATCH, 32, 0, stream>>>(lbph, gsums, tex38);
  k_gemm16<<<8, 32, 0, stream>>>(tex38, 38, W_tex, 128, b_tex, texf, 128, 38, 128, 1);

  // ---- fusion head
  k_comb<<<BATCH, 288, 0, stream>>>(cnn, qtn, texf, comb);
  k_gemm16<<<16, 32, 0, stream>>>(comb, 261, W_fus, 256, b_fus, fused, 256, 261, 256, 1);
  k_gemm16<<<1, 32, 0, stream>>>(fused, 256, W_fin, 5, b_fin, out, 5, 256, 5, 0);
}